// Correlation_71502615543889
// MI455X (gfx1250) — compile-verified
//
#include <hip/hip_runtime.h>

typedef __attribute__((ext_vector_type(2))) float v2f;
typedef __attribute__((ext_vector_type(8))) float v8f;

#define AS1 __attribute__((address_space(1)))
#define AS3 __attribute__((address_space(3)))

namespace {
constexpr int Bc  = 8;      // batch
constexpr int Cc  = 128;    // channels = K dimension
constexpr int Hc  = 128;
constexpr int Wc  = 256;
constexpr int HW  = Hc * Wc;
constexpr int KD  = 9;      // kernel diameter
constexpr int ND  = KD * KD;   // 81
constexpr int WPB = 8;      // waves per block (256 threads)
constexpr int XT  = WPB * 16;  // 128 output x per workgroup
constexpr int XH  = XT + 16;   // 144 staged x' (+-8 halo)
constexpr int CS  = 4;      // channels per stage (one K=4 WMMA step)
constexpr int CP  = CS + 2; // LDS channel pitch: keeps v2f 8B-aligned,
                            // bank stride 6 -> conflict-free 16-lane reads
constexpr int NST = Cc / CS;   // 32 stages
}

// Async global->LDS copy of one f32 (CDNA5 ASYNCcnt path), with portable
// fallback (global_load + ds_store) if the builtin is not declared.
__device__ __forceinline__ void cp_lds_f32(float* dst, const float* src) {
#if __has_builtin(__builtin_amdgcn_global_load_async_to_lds_b32)
  __builtin_amdgcn_global_load_async_to_lds_b32((AS1 int*)(float*)src,
                                                (AS3 int*)dst, 0, 0);
#else
  *dst = *src;
#endif
}

__device__ __forceinline__ void wait_cp() {
#if __has_builtin(__builtin_amdgcn_global_load_async_to_lds_b32)
#if __has_builtin(__builtin_amdgcn_s_wait_asynccnt)
  __builtin_amdgcn_s_wait_asynccnt(0);
#else
  asm volatile("s_wait_asynccnt 0x0" ::: "memory");
#endif
#endif
}

// Workgroup = 8 waves = half an image row of one (b, y).
// Wave w owns the 16-wide x tile at x0 + 16*w and computes all 81
// correlations for it: D = A x B, M = 16 x, N = 16 x', K = channels,
// via V_WMMA_F32_16X16X4_F32 (exact fp32). Two N tiles (x' bases xt-8,
// xt+8) cover the +-4 band. Zero padding lives in the LDS halo, so the
// hot loop has no predication at all (WMMA requires EXEC = all 1s anyway).
__global__ __launch_bounds__(WPB * 32) void corr_wmma_lds(
    const float* __restrict__ f1, const float* __restrict__ f2,
    float* __restrict__ out) {
  __shared__ float sF2[KD][XH][CP];   // 9*144*6*4  = 31104 B
  __shared__ float sF1[XT][CP];       // 128*6*4    =  3072 B
  __shared__ float sD[WPB][512];      // epilogue   = 16384 B

  const int tid  = threadIdx.x;
  const int lane = tid & 31;
  const int wave = tid >> 5;
  const int n    = lane & 15;   // N (and staged-x) index
  const int kHi  = lane >> 4;   // selects K pair {0,1} vs {2,3}
  const int xtl  = wave * 16;   // local x-tile base

  const int wg   = blockIdx.x;          // ((b*128 + y)*2 + half)
  const int half = wg & 1;
  const int y    = (wg >> 1) & (Hc - 1);
  const int b    = wg >> 8;
  const int x0   = half * XT;

  v8f acc[KD][2];
#pragma unroll
  for (int dy = 0; dy < KD; ++dy) { acc[dy][0] = {}; acc[dy][1] = {}; }

  const float* f2b = f2 + (size_t)b * Cc * HW;
  const float* f1b = f1 + (size_t)b * Cc * HW + (size_t)y * Wc + x0;

  for (int s = 0; s < NST; ++s) {
    const int c0 = s * CS;

    // ---- stage f2: 9 rows x 4 ch x 144 x', halo zero-filled in LDS
    for (int idx = tid; idx < KD * CS * XH; idx += WPB * 32) {
      const int xl = idx % XH;          // consecutive tids -> consecutive x'
      const int rc = idx / XH;
      const int c  = rc & (CS - 1);
      const int dy = rc >> 2;
      const int xg = x0 - 8 + xl;
      const int yg = y + dy - 4;
      float* dst = &sF2[dy][xl][c];
      if (((unsigned)xg < (unsigned)Wc) & ((unsigned)yg < (unsigned)Hc))
        cp_lds_f32(dst, f2b + ((size_t)(c0 + c) * Hc + yg) * Wc + xg);
      else
        *dst = 0.f;
    }
    // ---- stage f1: 4 ch x 128 x (always in range)
    for (int idx = tid; idx < CS * XT; idx += WPB * 32) {
      const int xl = idx & (XT - 1);
      const int c  = idx >> 7;
      cp_lds_f32(&sF1[xl][c], f1b + (size_t)(c0 + c) * HW + xl);
    }
    wait_cp();
    __syncthreads();

    // ---- compute: one K=4 step, 18 WMMAs, all operands from LDS,
    //      no predication, no EXEC churn.
    {
      const int p = 2 * kHi;   // even -> v2f reads are 8B aligned
      const v2f a = *(const v2f*)&sF1[xtl + n][p];
#pragma unroll
      for (int dy = 0; dy < KD; ++dy) {
        const v2f b0 = *(const v2f*)&sF2[dy][xtl + n][p];
        const v2f b1 = *(const v2f*)&sF2[dy][xtl + 16 + n][p];
        acc[dy][0] = __builtin_amdgcn_wmma_f32_16x16x4_f32(
            false, a, false, b0, (short)0, acc[dy][0], false, false);
        acc[dy][1] = __builtin_amdgcn_wmma_f32_16x16x4_f32(
            false, a, false, b1, (short)0, acc[dy][1], false, false);
      }
    }
    __syncthreads();   // before next stage overwrites sF1/sF2
  }

  // ---- epilogue: per dy, dump both D tiles into this wave's private LDS
  // slice (wave-internal DS ordering, no barrier needed), then re-read in
  // (kidx, x) order so global stores are dense 16-element runs.
  float* myD = &sD[wave][0];
  const float  scale = 1.0f / Cc;
  const size_t oB = (size_t)b * ND * HW + (size_t)y * Wc + (x0 + xtl);
#pragma unroll
  for (int dy = 0; dy < KD; ++dy) {
#pragma unroll
    for (int j = 0; j < 8; ++j) {
      const int m = j + 8 * kHi;            // D row held by this lane
      myD[m * 16 + n]       = acc[dy][0][j];
      myD[256 + m * 16 + n] = acc[dy][1][j];
    }
#pragma unroll
    for (int r = 0; r < 5; ++r) {
      const int idx = r * 32 + lane;        // (dxi, m) flat, 144 valid
      if (idx < 144) {
        const int dxi = idx >> 4;           // 0..8  -> dx = dxi - 4
        const int m   = idx & 15;
        const int s2  = m + dxi - 4;        // m + dx
        const int t   = (s2 >= 8) ? 1 : 0;  // which N tile holds it
        const int nn  = s2 + 8 - 16 * t;
        out[oB + (size_t)(dy * KD + dxi) * HW + m] =
            myD[t * 256 + m * 16 + nn] * scale;
      }
    }
  }
}

extern "C" void kernel_launch(void* const* d_in, const int* in_sizes, int n_in,
                              void* d_out, int out_size, void* d_ws,
                              size_t ws_size, hipStream_t stream) {
  (void)in_sizes; (void)n_in; (void)d_ws; (void)ws_size; (void)out_size;
  const float* f1  = (const float*)d_in[0];
  const float* f2  = (const float*)d_in[1];
  float*       out = (float*)d_out;

  // 8 batches * 128 rows * 2 half-rows = 2048 workgroups of 256 threads.
  const int blocks = Bc * Hc * 2;
  corr_wmma_lds<<<blocks, WPB * 32, 0, stream>>>(f1, f2, out);
}